// ALCOVECell_5918464934684
// MI455X (gfx1250) — compile-verified
//
#include <hip/hip_runtime.h>
#include <hip/hip_bf16.h>

// ALCOVE cell step for MI455X (gfx1250, wave32).
// B=256, D=64, H=2048, O=32. All fp32.
// Memory-bound on association (64MB read x2 + 64MB write); WMMA f32 16x16x4
// for the two distance GEMMs and the two attention-gradient GEMMs.
// Assoc streaming uses b128 loads/stores; new_assoc written non-temporally so
// the second assoc read stays resident in the 192MB L2.

#define B_SZ 256
#define D_SZ 64
#define H_SZ 2048
#define O_SZ 32

#define BETA  6.5f
#define PHI   2.0f
#define LAM_A 0.001f
#define LAM_W 0.003f
#define TWO_OVER_O 0.0625f   // 2/32

typedef __attribute__((ext_vector_type(2))) float v2f;
typedef __attribute__((ext_vector_type(4))) float v4f;
typedef __attribute__((ext_vector_type(8))) float v8f;

__device__ __forceinline__ v8f wmma_f32_16x16x4(v2f a, v2f b, v8f c) {
  // (neg_a, A, neg_b, B, c_mod, C, reuse_a, reuse_b)
  return __builtin_amdgcn_wmma_f32_16x16x4_f32(false, a, false, b, (short)0, c,
                                               false, false);
}

// ---------------------------------------------------------------------------
// Prep: az = att*z, A0[b] = sum_d att*z^2     (grid 256 x 64 threads)
// ---------------------------------------------------------------------------
__global__ void prep_b_kernel(const float* __restrict__ z,
                              const float* __restrict__ att,
                              float* __restrict__ az,
                              float* __restrict__ A0) {
  __shared__ float red[64];
  const int b = blockIdx.x, t = threadIdx.x;
  const float zv = z[b * D_SZ + t];
  const float av = att[b * D_SZ + t];
  az[b * D_SZ + t] = av * zv;
  red[t] = av * zv * zv;
  __syncthreads();
  if (t < 32) {
    float v = red[t] + red[t + 32];
    #pragma unroll
    for (int m = 16; m >= 1; m >>= 1) v += __shfl_xor(v, m, 32);
    if (t == 0) A0[b] = v;
  }
}

// c2 = coords^2 elementwise       (grid 128 x 256 threads, 131072 elems)
__global__ void prep_c2_kernel(const float* __restrict__ coords,
                               float* __restrict__ c2) {
  const int i = blockIdx.x * 256 + threadIdx.x;
  const v4f v = ((const v4f*)coords)[i];
  ((v4f*)c2)[i] = v * v;
}

// ---------------------------------------------------------------------------
// Distance GEMMs via WMMA f32 16x16x4:
//   U = (att*z) @ C^T,  V = att @ (C^2)^T   (M=256, N=2048, K=64)
//   q = A0 - 2U + V;  s = exp(-beta*sqrt(q));  w = beta*s/(2*sqrt(q))
// One wave per 16x16 (b,h) tile; 2048 tiles; 8 waves/block -> 256 blocks.
// ---------------------------------------------------------------------------
__global__ void dist_wmma_kernel(const float* __restrict__ az,
                                 const float* __restrict__ att,
                                 const float* __restrict__ coords,
                                 const float* __restrict__ c2,
                                 const float* __restrict__ A0,
                                 float* __restrict__ s_out,
                                 float* __restrict__ w_out) {
  const int wave = threadIdx.x >> 5;
  const int lane = threadIdx.x & 31;
  const int tile = blockIdx.x * 8 + wave;       // 0..2047
  const int b0 = (tile >> 7) * 16;              // 16 b-tiles
  const int h0 = (tile & 127) * 16;             // 128 h-tiles
  const int mn = lane & 15;                     // M for A rows, N for B cols
  const int khalf = (lane >> 4) << 1;           // 0 or 2

  const float* aRow1 = az  + (b0 + mn) * D_SZ;
  const float* aRow2 = att + (b0 + mn) * D_SZ;
  const float* bRow1 = coords + (h0 + mn) * D_SZ;
  const float* bRow2 = c2     + (h0 + mn) * D_SZ;

  v8f accU = {}; v8f accV = {};
  #pragma unroll
  for (int kk = 0; kk < 16; ++kk) {
    const int k = kk * 4 + khalf;
    v2f a1; a1.x = aRow1[k]; a1.y = aRow1[k + 1];
    v2f b1; b1.x = bRow1[k]; b1.y = bRow1[k + 1];
    v2f a2; a2.x = aRow2[k]; a2.y = aRow2[k + 1];
    v2f b2; b2.x = bRow2[k]; b2.y = bRow2[k + 1];
    accU = wmma_f32_16x16x4(a1, b1, accU);
    accV = wmma_f32_16x16x4(a2, b2, accV);
  }

  const int half = lane >> 4;
  #pragma unroll
  for (int r = 0; r < 8; ++r) {
    const int m = r + 8 * half;       // C/D layout: vgpr r, lanes>=16 -> M+8
    const int b = b0 + m, h = h0 + mn;
    float q = A0[b] - 2.0f * accU[r] + accV[r];
    q = fmaxf(q, 0.0f);
    const float dd = sqrtf(q);
    const float sv = __expf(-BETA * dd);
    const float wv = (dd > 1e-20f) ? (BETA * 0.5f * sv / dd) : 0.0f;
    s_out[b * H_SZ + h] = sv;
    w_out[b * H_SZ + h] = wv;
  }
}

// ---------------------------------------------------------------------------
// x_out[b,o] = sum_h s[b,h]*assoc[b,h,o]; then teacher/grad g[b,o].
// grid = 256 (b), 256 threads. Each thread: o-group (4 o's), h-stride 32.
// b128 loads; streams assoc once (stays in L2 for pass 2).
// ---------------------------------------------------------------------------
__global__ void xout_kernel(const float* __restrict__ assoc,
                            const float* __restrict__ s,
                            const float* __restrict__ onehot,
                            float* __restrict__ xout_scaled,
                            float* __restrict__ g_ws) {
  __shared__ v4f red[256];
  const int b = blockIdx.x;
  const int t = threadIdx.x;
  const int og = t & 7;          // o = og*4 .. og*4+3
  const int hg = t >> 3;         // 0..31
  const float* ab = assoc + (size_t)b * (H_SZ * O_SZ);
  const float* sb = s + b * H_SZ;
  v4f acc = {};
  for (int h = hg; h < H_SZ; h += 32) {
    const v4f a = *(const v4f*)(ab + h * O_SZ + og * 4);
    acc += sb[h] * a;
  }
  red[t] = acc;
  __syncthreads();
  if (t < 128) red[t] += red[t + 128];
  __syncthreads();
  if (t < 64) red[t] += red[t + 64];
  __syncthreads();
  if (t < 32) red[t] += red[t + 32];
  __syncthreads();
  if (t < 16) red[t] += red[t + 16];
  __syncthreads();
  if (t < 8) {
    const v4f x4 = red[t] + red[t + 8];
    #pragma unroll
    for (int c = 0; c < 4; ++c) {
      const float x = x4[c];
      const int o = t * 4 + c;
      const float oh = onehot[b * O_SZ + o];
      const float tmin = fminf(-1.0f, x);
      const float tmax = fmaxf(1.0f, x);
      const float teacher = tmin - oh * tmin + oh * tmax;
      const float resid = teacher - x;
      const float dteach = (1.0f - oh) * ((x < -1.0f) ? 1.0f : 0.0f)
                         + oh * ((x > 1.0f) ? 1.0f : 0.0f);
      const float g = TWO_OVER_O * resid * (dteach - 1.0f);
      xout_scaled[b * O_SZ + o] = PHI * x;
      g_ws[b * O_SZ + o] = g;
    }
  }
}

// ---------------------------------------------------------------------------
// new_assoc = assoc - LAM_W*g[b,o]*s[b,h];  gs[b,h]=sum_o g*assoc;
// coef[b,h] = -gs*w.  grid (256,16) x 256 threads.
// Lane covers 4 o's (b128), 4 h's per wave-iteration; 3-shuffle reduce per h.
// new_assoc stored non-temporally (write-once, never re-read).
// ---------------------------------------------------------------------------
__global__ void assoc_update_kernel(const float* __restrict__ assoc,
                                    const float* __restrict__ s,
                                    const float* __restrict__ w,
                                    const float* __restrict__ g_ws,
                                    float* __restrict__ new_assoc,
                                    float* __restrict__ coef) {
  const int b = blockIdx.x;
  const int hbase = blockIdx.y * 128;
  const int lane = threadIdx.x & 31;
  const int wave = threadIdx.x >> 5;
  const int og = lane & 7;       // o = og*4 .. og*4+3
  const int hs = lane >> 3;      // 0..3
  const v4f g4 = *(const v4f*)(g_ws + b * O_SZ + og * 4);
  const size_t base = (size_t)b * (H_SZ * O_SZ);
  #pragma unroll
  for (int j = 0; j < 4; ++j) {
    const int h = hbase + wave * 16 + j * 4 + hs;
    const float sv = s[b * H_SZ + h];
    const size_t idx = base + (size_t)h * O_SZ + og * 4;
    const v4f a = *(const v4f*)(assoc + idx);
    const v4f na = a - (LAM_W * sv) * g4;
    __builtin_nontemporal_store(na, (v4f*)(new_assoc + idx));
    float p = fmaf(g4.x, a.x, fmaf(g4.y, a.y, fmaf(g4.z, a.z, g4.w * a.w)));
    p += __shfl_xor(p, 1, 32);
    p += __shfl_xor(p, 2, 32);
    p += __shfl_xor(p, 4, 32);
    if (og == 0) coef[b * H_SZ + h] = -p * w[b * H_SZ + h];
  }
}

// ---------------------------------------------------------------------------
// Attention-grad GEMMs via WMMA: P1 = coef@C, P2 = coef@C2
// (M=256, N=64, K=2048). 64 tiles, 8 waves/block -> 8 blocks.
// ---------------------------------------------------------------------------
__global__ void attgrad_wmma_kernel(const float* __restrict__ coef,
                                    const float* __restrict__ coords,
                                    const float* __restrict__ c2,
                                    float* __restrict__ P1,
                                    float* __restrict__ P2) {
  const int wave = threadIdx.x >> 5;
  const int lane = threadIdx.x & 31;
  const int tile = blockIdx.x * 8 + wave;       // 0..63
  const int b0 = (tile >> 2) * 16;              // 16 b-tiles
  const int d0 = (tile & 3) * 16;               // 4 d-tiles
  const int mn = lane & 15;
  const int khalf = (lane >> 4) << 1;

  const float* aRow = coef + (b0 + mn) * H_SZ;
  v8f acc1 = {}; v8f acc2 = {};
  for (int kk = 0; kk < H_SZ / 4; ++kk) {
    const int k = kk * 4 + khalf;
    v2f a;  a.x  = aRow[k];                    a.y  = aRow[k + 1];
    v2f b1; b1.x = coords[k * D_SZ + d0 + mn]; b1.y = coords[(k + 1) * D_SZ + d0 + mn];
    v2f b2; b2.x = c2[k * D_SZ + d0 + mn];     b2.y = c2[(k + 1) * D_SZ + d0 + mn];
    acc1 = wmma_f32_16x16x4(a, b1, acc1);
    acc2 = wmma_f32_16x16x4(a, b2, acc2);
  }
  const int half = lane >> 4;
  #pragma unroll
  for (int r = 0; r < 8; ++r) {
    const int m = r + 8 * half;
    P1[(b0 + m) * D_SZ + d0 + mn] = acc1[r];
    P2[(b0 + m) * D_SZ + d0 + mn] = acc2[r];
  }
}

// ---------------------------------------------------------------------------
// new_att[b,d] = max(att - LAM_A*(z^2*S0 - 2z*P1 + P2), 0), S0[b]=sum_h coef.
// grid 256 x 64 threads.
// ---------------------------------------------------------------------------
__global__ void attention_kernel(const float* __restrict__ coef,
                                 const float* __restrict__ z,
                                 const float* __restrict__ att,
                                 const float* __restrict__ P1,
                                 const float* __restrict__ P2,
                                 float* __restrict__ new_att) {
  __shared__ float red[64];
  const int b = blockIdx.x, t = threadIdx.x;
  const float* cb = coef + b * H_SZ;
  float acc = 0.0f;
  for (int h = t; h < H_SZ; h += 64) acc += cb[h];
  red[t] = acc;
  __syncthreads();
  if (t < 32) {
    float v = red[t] + red[t + 32];
    #pragma unroll
    for (int m = 16; m >= 1; m >>= 1) v += __shfl_xor(v, m, 32);
    if (t == 0) red[0] = v;
  }
  __syncthreads();
  const float S0 = red[0];
  const int i = b * D_SZ + t;
  const float zv = z[i];
  const float dl = zv * zv * S0 - 2.0f * zv * P1[i] + P2[i];
  new_att[i] = fmaxf(att[i] - LAM_A * dl, 0.0f);
}

// ---------------------------------------------------------------------------
// Host launch
// ---------------------------------------------------------------------------
extern "C" void kernel_launch(void* const* d_in, const int* in_sizes, int n_in,
                              void* d_out, int out_size, void* d_ws, size_t ws_size,
                              hipStream_t stream) {
  (void)in_sizes; (void)n_in; (void)out_size; (void)ws_size;
  const float* z      = (const float*)d_in[0];   // (B,D)
  const float* onehot = (const float*)d_in[1];   // (B,O)
  const float* att    = (const float*)d_in[2];   // (B,D)
  const float* assoc  = (const float*)d_in[3];   // (B,H,O)
  const float* coords = (const float*)d_in[4];   // (H,D)

  float* out = (float*)d_out;
  float* xout_scaled = out;                              // B*O = 8192
  float* new_att     = out + B_SZ * O_SZ;                // B*D = 16384
  float* new_assoc   = out + B_SZ * O_SZ + B_SZ * D_SZ;  // B*H*O

  float* ws = (float*)d_ws;
  float* s_buf = ws;                       size_t off = (size_t)B_SZ * H_SZ;
  float* w_buf = ws + off;                 off += (size_t)B_SZ * H_SZ;
  float* coef  = ws + off;                 off += (size_t)B_SZ * H_SZ;
  float* g_ws  = ws + off;                 off += (size_t)B_SZ * O_SZ;
  float* az    = ws + off;                 off += (size_t)B_SZ * D_SZ;
  float* c2    = ws + off;                 off += (size_t)H_SZ * D_SZ;
  float* A0    = ws + off;                 off += (size_t)B_SZ;
  float* P1    = ws + off;                 off += (size_t)B_SZ * D_SZ;
  float* P2    = ws + off;

  prep_b_kernel<<<B_SZ, D_SZ, 0, stream>>>(z, att, az, A0);
  prep_c2_kernel<<<(H_SZ * D_SZ) / 1024, 256, 0, stream>>>(coords, c2);
  dist_wmma_kernel<<<256, 256, 0, stream>>>(az, att, coords, c2, A0, s_buf, w_buf);
  xout_kernel<<<B_SZ, 256, 0, stream>>>(assoc, s_buf, onehot, xout_scaled, g_ws);
  dim3 gC(B_SZ, 16);
  assoc_update_kernel<<<gC, 256, 0, stream>>>(assoc, s_buf, w_buf, g_ws,
                                              new_assoc, coef);
  attgrad_wmma_kernel<<<8, 256, 0, stream>>>(coef, coords, c2, P1, P2);
  attention_kernel<<<B_SZ, D_SZ, 0, stream>>>(coef, z, att, P1, P2, new_att);
}